// KWinner_44538810859565
// MI455X (gfx1250) — compile-verified
//
#include <hip/hip_runtime.h>
#include <cstdint>

// ---------------------------------------------------------------------------
// k-winners-take-all, one row (4096 cols) per 256-thread block.
// Memory-bound (256 MiB traffic -> ~11 us at 23.3 TB/s): stage row to LDS with
// CDNA5 async global->LDS DMA, radix-select the k-th largest boosted value,
// write masked x.
// ---------------------------------------------------------------------------

#define N_COLS 4096
#define TPB    256
#define EPT    (N_COLS / TPB)   // 16 elements per thread

#if defined(__has_builtin)
#  if __has_builtin(__builtin_amdgcn_global_load_async_to_lds_b128)
#    define HAVE_ASYNC_LDS 1
#  endif
#  if __has_builtin(__builtin_amdgcn_s_wait_asynccnt)
#    define HAVE_WAIT_ASYNC 1
#  endif
#endif
#ifndef HAVE_ASYNC_LDS
#  define HAVE_ASYNC_LDS 0
#endif
#ifndef HAVE_WAIT_ASYNC
#  define HAVE_WAIT_ASYNC 0
#endif

typedef __attribute__((ext_vector_type(4))) int v4i;
typedef __attribute__((address_space(1))) v4i gv4i;   // global v4i
typedef __attribute__((address_space(3))) v4i lv4i;   // LDS v4i

__global__ __launch_bounds__(TPB) void kwinner_kernel(
    const float* __restrict__ x,
    const float* __restrict__ duty,
    const int*   __restrict__ kptr,
    float*       __restrict__ out,
    int n)
{
    __shared__ __align__(16) float    lds_x[N_COLS];   // async DMA landing
    __shared__ unsigned               hist[256];
    __shared__ unsigned               s_state[2];      // {prefix, k_remaining}

    const int       tid  = threadIdx.x;
    const long long row  = blockIdx.x;
    const float*    xrow = x + row * (long long)N_COLS;

    // ---- stage row into LDS ------------------------------------------------
#if HAVE_ASYNC_LDS
    #pragma unroll
    for (int c = 0; c < 4; ++c) {
        const int idx = tid + c * TPB;                 // float4 chunk index
        __builtin_amdgcn_global_load_async_to_lds_b128(
            (gv4i*)(xrow + 4 * idx),
            (lv4i*)(&lds_x[4 * idx]),
            0, 0);
    }
#else
    #pragma unroll
    for (int c = 0; c < 4; ++c) {
        const int idx = tid + c * TPB;
        ((float4*)lds_x)[idx] = ((const float4*)xrow)[idx];
    }
#endif

    const int   k      = kptr[0];
    const float target = (float)k / (float)n;

#if HAVE_ASYNC_LDS
#  if HAVE_WAIT_ASYNC
    __builtin_amdgcn_s_wait_asynccnt(0);
#  else
    asm volatile("s_wait_asynccnt 0" ::: "memory");
#  endif
#endif
    __syncthreads();

    // ---- boosted values -> monotone u32 keys (regs) ------------------------
    float    xr[EPT];
    unsigned ur[EPT];
    #pragma unroll
    for (int i = 0; i < EPT; ++i) {
        const int   j  = tid + i * TPB;                // coalesced / no bank conflicts
        const float xv = lds_x[j];
        const float bv = xv * __expf(target - duty[j]); // boostStrength == 1.0
        unsigned u = __float_as_uint(bv);
        u = (u & 0x80000000u) ? ~u : (u | 0x80000000u); // larger float => larger u
        xr[i] = xv;
        ur[i] = u;
    }

    if (tid == 0) { s_state[0] = 0u; s_state[1] = (unsigned)k; }
    __syncthreads();

    // ---- 4-pass byte radix select for the k-th largest key ------------------
    for (int p = 3; p >= 0; --p) {
        hist[tid] = 0u;
        __syncthreads();

        const unsigned prefix  = s_state[0];
        const unsigned krem    = s_state[1];
        const int      shift   = p * 8;
        const unsigned mask_hi = (p == 3) ? 0u : (0xFFFFFFFFu << ((p + 1) * 8));

        #pragma unroll
        for (int i = 0; i < EPT; ++i) {
            const unsigned u = ur[i];
            if ((u & mask_hi) == prefix)
                atomicAdd(&hist[(u >> shift) & 0xFFu], 1u);
        }
        __syncthreads();

        // inclusive suffix sum over 256 bins (Hillis-Steele, 8 steps)
        #pragma unroll
        for (int off = 1; off < 256; off <<= 1) {
            const unsigned add = (tid + off < 256) ? hist[tid + off] : 0u;
            __syncthreads();
            hist[tid] += add;
            __syncthreads();
        }

        const unsigned sfx       = hist[tid];                        // count in bins >= tid
        const unsigned sfx_above = (tid < 255) ? hist[tid + 1] : 0u; // count in bins >  tid
        if (sfx >= krem && sfx_above < krem) {                       // exactly one thread
            s_state[0] = prefix | ((unsigned)tid << shift);
            s_state[1] = krem - sfx_above;
        }
        __syncthreads();
    }

    // ---- mask & store (threshold compare matches ref's `boosted >= bottom`) -
    const unsigned thresh = s_state[0];
    float* orow = out + row * (long long)N_COLS;
    #pragma unroll
    for (int i = 0; i < EPT; ++i) {
        const int j = tid + i * TPB;
        orow[j] = (ur[i] >= thresh) ? xr[i] : 0.0f;
    }
}

extern "C" void kernel_launch(void* const* d_in, const int* in_sizes, int n_in,
                              void* d_out, int out_size, void* d_ws, size_t ws_size,
                              hipStream_t stream) {
    (void)n_in; (void)d_ws; (void)ws_size; (void)out_size;
    const float* x    = (const float*)d_in[0];
    const float* duty = (const float*)d_in[1];
    const int*   kptr = (const int*)d_in[2];
    float*       out  = (float*)d_out;

    const int n    = in_sizes[1];                 // 4096
    const int rows = in_sizes[0] / n;             // 8192

    if (n == N_COLS && rows > 0) {
        kwinner_kernel<<<dim3(rows), dim3(TPB), 0, stream>>>(x, duty, kptr, out, n);
    }
}